// MFilm_27822798144211
// MI455X (gfx1250) — compile-verified
//
#include <hip/hip_runtime.h>
#include <hip/hip_bf16.h>

// ---------------------------------------------------------------------------
// MI455X (gfx1250) implementation of the Mamba-FiLM reference.
//
// Roofline: output write = 512 MiB fp32 @ 23.3 TB/s ~ 22us -> store-bound.
// The sequential Mamba scan (L=8192) is parallelized as a chunked linear
// recurrence (pass1 chunk summaries -> serial combine -> pass2 replay).
// The final MLP (K=4) maps exactly onto V_WMMA_F32_16X16X4_F32, computed as
// D = W^T * h^T so the D-matrix lane layout (N = time across lanes) writes
// the transposed gamma/beta layout with coalesced 64B runs; stores use
// non-temporal hints so the 512 MiB stream doesn't thrash L2.
// ---------------------------------------------------------------------------

typedef float v2f __attribute__((ext_vector_type(2)));
typedef float v8f __attribute__((ext_vector_type(8)));

#define L_SEQ   8192
#define BATCH_N 16
#define NCHUNK  128
#define CHUNK   64      // NCHUNK * CHUNK == L_SEQ

struct LayerPtrs {
  const float *norm_w, *norm_b, *in_proj, *conv_w, *conv_b, *x_proj,
              *dt_w, *dt_b, *A_log, *D, *out_proj;
};

__device__ __forceinline__ float sigmoid_f(float x) { return 1.0f / (1.0f + __expf(-x)); }
__device__ __forceinline__ float softplus_f(float x) { return (x > 20.0f) ? x : __logf(1.0f + __expf(x)); }

// ---------------------------------------------------------------------------
// Kernel 1: max-pool (64-wide) + dense (1 -> 2).  h0[b, t, 0:2]
// ---------------------------------------------------------------------------
__global__ void pool_dense_kernel(const float* __restrict__ x,
                                  const float* __restrict__ dw,
                                  const float* __restrict__ db,
                                  float* __restrict__ h0)
{
  int o = blockIdx.x * blockDim.x + threadIdx.x;      // 0 .. 131071
  const float4* p = (const float4*)(x + (size_t)o * 64);
  float m = -3.402823466e38f;
#pragma unroll
  for (int i = 0; i < 16; i++) {
    float4 v = p[i];
    m = fmaxf(m, fmaxf(fmaxf(v.x, v.y), fmaxf(v.z, v.w)));
  }
  h0[(size_t)o * 2 + 0] = m * dw[0] + db[0];
  h0[(size_t)o * 2 + 1] = m * dw[1] + db[1];
}

// ---------------------------------------------------------------------------
// Kernels 2/4: chunked Mamba scan.  One thread per (dir, b, chunk).
// Pass1: h_in = 0, record (prod dA, local final h) per chunk.
// Pass2: h_in from combine, emit mixer output + residual into m[b,t,4].
// Layer params staged in LDS (dir is block-uniform).
// ---------------------------------------------------------------------------
template <bool PASS2>
__global__ void mamba_scan_kernel(const float* __restrict__ h0,
                                  float* __restrict__ sumA,
                                  float* __restrict__ sumH,
                                  const float* __restrict__ hinit,
                                  float* __restrict__ mout,
                                  LayerPtrs fwd, LayerPtrs bwd)
{
  __shared__ float P[256];
  const int tid = threadIdx.x;
  const int idx = blockIdx.x * 256 + tid;             // 0 .. 4095
  const int dir = idx >> 11;                          // block-uniform (2048/256)
  const int b   = (idx >> 7) & (BATCH_N - 1);
  const int c   = idx & (NCHUNK - 1);
  const LayerPtrs lp = dir ? bwd : fwd;

  {
    const float* srcs[11] = { lp.norm_w, lp.norm_b, lp.in_proj, lp.conv_w, lp.conv_b,
                              lp.x_proj, lp.dt_w, lp.dt_b, lp.A_log, lp.D, lp.out_proj };
    const int sizes[11] = { 2, 2, 16, 16, 4, 132, 4, 4, 64, 4, 8 };
    int off = 0;
#pragma unroll
    for (int a = 0; a < 11; a++) { if (tid < sizes[a]) P[off + tid] = srcs[a][tid]; off += sizes[a]; }
  }
  __syncthreads();
  if (tid < 64) P[180 + tid] = -__expf(P[180 + tid]);   // A = -exp(A_log)
  __syncthreads();

  const float NW0 = P[0], NW1 = P[1], NB0 = P[2], NB1 = P[3];
  const float* IP   = &P[4];     // in_proj (2,8)
  const float* CW   = &P[20];    // conv_w (4,4)
  const float* CB   = &P[36];
  const float* XP   = &P[40];    // x_proj (4,33)
  const float* DTW  = &P[172];
  const float* DTB  = &P[176];
  const float* ANEG = &P[180];   // (4,16)
  const float* DP   = &P[244];
  const float* OP   = &P[248];   // out_proj (4,2)

  // recompute layernorm + in_proj for local (direction-mapped) time t
  auto inproj = [&](int t, float xr[4], float zz[4], float* r0, float* r1) {
    int tau = dir ? (L_SEQ - 1 - t) : t;
    float x0 = h0[((size_t)b * L_SEQ + tau) * 2 + 0];
    float x1 = h0[((size_t)b * L_SEQ + tau) * 2 + 1];
    float mu = 0.5f * (x0 + x1);
    float d0 = x0 - mu, d1 = x1 - mu;
    float inv = rsqrtf(0.5f * (d0 * d0 + d1 * d1) + 1e-5f);
    float xn0 = d0 * inv * NW0 + NB0;
    float xn1 = d1 * inv * NW1 + NB1;
#pragma unroll
    for (int j = 0; j < 4; j++) xr[j] = xn0 * IP[j] + xn1 * IP[8 + j];
    if (zz) {
#pragma unroll
      for (int j = 0; j < 4; j++) zz[j] = xn0 * IP[4 + j] + xn1 * IP[12 + j];
    }
    if (r0) { *r0 = x0; *r1 = x1; }
  };

  const int t0 = c * CHUNK;

  float hst[64];
  float Ap[64];
  if (PASS2) {
    size_t base = ((size_t)(dir * BATCH_N + b) * NCHUNK + c) * 64;
#pragma unroll
    for (int i = 0; i < 64; i++) hst[i] = hinit[base + i];
  } else {
#pragma unroll
    for (int i = 0; i < 64; i++) { hst[i] = 0.0f; Ap[i] = 1.0f; }
  }

  // conv lookback: raw in_proj x at t0-3 .. t0-1 (zero-padded before t=0)
  float hist[3][4];
#pragma unroll
  for (int k = 0; k < 3; k++) {
    int t = t0 - 3 + k;
    if (t < 0) { hist[k][0] = hist[k][1] = hist[k][2] = hist[k][3] = 0.0f; }
    else       { inproj(t, hist[k], nullptr, nullptr, nullptr); }
  }

  for (int tt = 0; tt < CHUNK; tt++) {
    int t = t0 + tt;
    float xr[4], zz[4], r0, r1;
    inproj(t, xr, zz, &r0, &r1);

    float xc[4];
#pragma unroll
    for (int d = 0; d < 4; d++) {
      float v = hist[0][d] * CW[d * 4 + 0] + hist[1][d] * CW[d * 4 + 1]
              + hist[2][d] * CW[d * 4 + 2] + xr[d] * CW[d * 4 + 3] + CB[d];
      xc[d] = v * sigmoid_f(v);                         // silu
    }
#pragma unroll
    for (int d = 0; d < 4; d++) { hist[0][d] = hist[1][d]; hist[1][d] = hist[2][d]; hist[2][d] = xr[d]; }

    float dtr = xc[0] * XP[0] + xc[1] * XP[33] + xc[2] * XP[66] + xc[3] * XP[99];
    float Bv[16], Cv[16];
#pragma unroll
    for (int s = 0; s < 16; s++) {
      Bv[s] = xc[0] * XP[1 + s]  + xc[1] * XP[34 + s]  + xc[2] * XP[67 + s]  + xc[3] * XP[100 + s];
      Cv[s] = xc[0] * XP[17 + s] + xc[1] * XP[50 + s]  + xc[2] * XP[83 + s]  + xc[3] * XP[116 + s];
    }
    float dt[4];
#pragma unroll
    for (int d = 0; d < 4; d++) dt[d] = softplus_f(dtr * DTW[d] + DTB[d]);

#pragma unroll
    for (int d = 0; d < 4; d++) {
      float dx = dt[d] * xc[d];
#pragma unroll
      for (int s = 0; s < 16; s++) {
        int i = d * 16 + s;
        float dA = __expf(dt[d] * ANEG[i]);
        if (!PASS2) Ap[i] *= dA;
        hst[i] = dA * hst[i] + dx * Bv[s];
      }
    }

    if (PASS2) {
      float o0 = r0, o1 = r1;                           // residual
#pragma unroll
      for (int d = 0; d < 4; d++) {
        float y = xc[d] * DP[d];
#pragma unroll
        for (int s = 0; s < 16; s++) y += hst[d * 16 + s] * Cv[s];
        y *= zz[d] * sigmoid_f(zz[d]);
        o0 += y * OP[d * 2 + 0];
        o1 += y * OP[d * 2 + 1];
      }
      int tau = dir ? (L_SEQ - 1 - t) : t;
      size_t mo = ((size_t)b * L_SEQ + tau) * 4 + dir * 2;
      mout[mo + 0] = o0;
      mout[mo + 1] = o1;
    }
  }

  if (!PASS2) {
    size_t base = ((size_t)(dir * BATCH_N + b) * NCHUNK + c) * 64;
#pragma unroll
    for (int i = 0; i < 64; i++) { sumA[base + i] = Ap[i]; sumH[base + i] = hst[i]; }
  }
}

// ---------------------------------------------------------------------------
// Kernel 3: serial combine of chunk summaries -> per-chunk initial states.
// 32 blocks (dir*16+b) x 64 threads (state d*16+s); 128-deep serial loop.
// ---------------------------------------------------------------------------
__global__ void combine_kernel(const float* __restrict__ sumA,
                               const float* __restrict__ sumH,
                               float* __restrict__ hinit)
{
  int g = blockIdx.x;          // dir*BATCH_N + b
  int sid = threadIdx.x;       // 0..63
  float run = 0.0f;
  for (int c = 0; c < NCHUNK; c++) {
    size_t base = ((size_t)g * NCHUNK + c) * 64 + sid;
    hinit[base] = run;
    run = sumA[base] * run + sumH[base];
  }
}

// ---------------------------------------------------------------------------
// Kernel 5: MLP (4 -> 1024) via V_WMMA_F32_16X16X4_F32, fused transpose/split.
// D = W^T (M=channels) x h^T (N=time), K=4.  One wave per (b, 16-time tile);
// B (the h tile) stays in registers across all 64 channel tiles; bias is
// folded into the C operand.  D layout: VGPR i -> M = i + 8*half, N = lane&15,
// so each per-VGPR store writes 16 consecutive time samples per half-wave.
// ---------------------------------------------------------------------------
__global__ void mlp_wmma_kernel(const float* __restrict__ m,
                                const float* __restrict__ W,     // (4,1024)
                                const float* __restrict__ bias,  // (1024)
                                float* __restrict__ out)         // gamma | beta
{
  int wave  = (blockIdx.x * blockDim.x + threadIdx.x) >> 5;  // 0 .. 8191
  int lane  = threadIdx.x & 31;
  int b     = wave >> 9;                                     // / 512
  int ttile = wave & 511;
  int t0    = ttile * 16;
  int half  = lane >> 4;
  int n     = lane & 15;
  int t     = t0 + n;

  // B tile = h^T (4 x 16): lanes 0-15 hold K rows 0,1; lanes 16-31 hold 2,3
  const float* mrow = m + ((size_t)b * L_SEQ + t) * 4;
  v2f bv;
  bv.x = mrow[half * 2 + 0];
  bv.y = mrow[half * 2 + 1];

  const size_t GAMMA_ELEMS = (size_t)BATCH_N * 512 * L_SEQ;  // 67108864

  for (int ct = 0; ct < 64; ct++) {
    int m0 = ct * 16;
    // A tile = W^T (16 x 4): lane row M = n, lanes 0-15 K=0,1; 16-31 K=2,3
    int k0 = half * 2;
    v2f av;
    av.x = W[(size_t)(k0 + 0) * 1024 + m0 + n];
    av.y = W[(size_t)(k0 + 1) * 1024 + m0 + n];
    // C = bias broadcast along N (per-lane channel = m0 + 8*half + i)
    v8f cv;
#pragma unroll
    for (int i = 0; i < 8; i++) cv[i] = bias[m0 + half * 8 + i];

    v8f dv = __builtin_amdgcn_wmma_f32_16x16x4_f32(
        /*neg_a=*/false, av, /*neg_b=*/false, bv,
        /*c_mod=*/(short)0, cv, /*reuse_a=*/false, /*reuse_b=*/false);

    // transposed store: out[b, ch, t] with gamma = ch<512, beta = ch>=512
    float* base;
    if (m0 < 512) base = out + ((size_t)b * 512 + m0) * L_SEQ;
    else          base = out + GAMMA_ELEMS + ((size_t)b * 512 + (m0 - 512)) * L_SEQ;
#pragma unroll
    for (int i = 0; i < 8; i++) {
      float* p = base + (size_t)(half * 8 + i) * L_SEQ + t;
      __builtin_nontemporal_store(dv[i], p);   // 512 MiB stream: NT, spare L2
    }
  }
}

// ---------------------------------------------------------------------------
extern "C" void kernel_launch(void* const* d_in, const int* in_sizes, int n_in,
                              void* d_out, int out_size, void* d_ws, size_t ws_size,
                              hipStream_t stream)
{
  (void)in_sizes; (void)n_in; (void)out_size; (void)ws_size;

  const float* x       = (const float*)d_in[0];
  const float* dense_w = (const float*)d_in[1];
  const float* dense_b = (const float*)d_in[2];
  const float* mlp_w   = (const float*)d_in[3];
  const float* mlp_b   = (const float*)d_in[4];

  LayerPtrs fwd = { (const float*)d_in[5],  (const float*)d_in[6],  (const float*)d_in[7],
                    (const float*)d_in[8],  (const float*)d_in[9],  (const float*)d_in[10],
                    (const float*)d_in[11], (const float*)d_in[12], (const float*)d_in[13],
                    (const float*)d_in[14], (const float*)d_in[15] };
  LayerPtrs bwd = { (const float*)d_in[16], (const float*)d_in[17], (const float*)d_in[18],
                    (const float*)d_in[19], (const float*)d_in[20], (const float*)d_in[21],
                    (const float*)d_in[22], (const float*)d_in[23], (const float*)d_in[24],
                    (const float*)d_in[25], (const float*)d_in[26] };

  float* ws    = (float*)d_ws;
  float* h0    = ws;                 //  262144 floats: (16, 8192, 2)
  float* sumA  = ws + 262144;        //  262144 floats: (32, 128, 64)
  float* sumH  = ws + 524288;        //  262144
  float* hinit = ws + 786432;        //  262144
  float* mbuf  = ws + 1048576;       //  524288 floats: (16, 8192, 4)
  float* out   = (float*)d_out;      //  134217728 floats: gamma | beta

  pool_dense_kernel<<<dim3(512), dim3(256), 0, stream>>>(x, dense_w, dense_b, h0);
  mamba_scan_kernel<false><<<dim3(16), dim3(256), 0, stream>>>(h0, sumA, sumH, hinit, mbuf, fwd, bwd);
  combine_kernel<<<dim3(32), dim3(64), 0, stream>>>(sumA, sumH, hinit);
  mamba_scan_kernel<true><<<dim3(16), dim3(256), 0, stream>>>(h0, sumA, sumH, hinit, mbuf, fwd, bwd);
  mlp_wmma_kernel<<<dim3(1024), dim3(256), 0, stream>>>(mbuf, mlp_w, mlp_b, out);
}